// MaskingAllocation_858993459835
// MI455X (gfx1250) — compile-verified
//
#include <hip/hip_runtime.h>

typedef float v2f __attribute__((ext_vector_type(2)));
typedef float v8f __attribute__((ext_vector_type(8)));
typedef int v4i __attribute__((vector_size(16)));  // b128 payload type for async LDS builtins

constexpr int B = 8, S = 2048, T = 2048, D = 128;
constexpr int LDSW = 132;  // 128 + 4 pad: conflict-free b64 reads, 16B-aligned b128 stores

#if defined(__HIP_DEVICE_COMPILE__) && \
    __has_builtin(__builtin_amdgcn_global_load_async_to_lds_b128)
#define MA_ASYNC_LDS 1
#else
#define MA_ASYNC_LDS 0
#endif

__device__ __forceinline__ void ma_wait_async() {
#if MA_ASYNC_LDS
#if __has_builtin(__builtin_amdgcn_s_wait_asynccnt)
  __builtin_amdgcn_s_wait_asynccnt(0);
#else
  asm volatile("s_wait_asynccnt 0x0" ::: "memory");
#endif
#endif
}

// Stage one 16x128 float tile (8 KB) into LDS; 128 threads, 16 floats each.
__device__ __forceinline__ void ma_stage_tile(float* ldsbase, const float* gsrc, int tid) {
  const int row = tid >> 3;       // 16 rows, 8 threads per row
  const int c0 = (tid & 7) * 16;  // 16 consecutive floats per thread
  const float* g = gsrc + (size_t)row * D + c0;
  float* l = ldsbase + row * LDSW + c0;
#if MA_ASYNC_LDS
  // global -> LDS direct, tracked by ASYNCcnt; same imm offset applies to both sides.
  __attribute__((address_space(1))) v4i* gp =
      (__attribute__((address_space(1))) v4i*)(unsigned long long)(size_t)g;
  __attribute__((address_space(3))) v4i* lp =
      (__attribute__((address_space(3))) v4i*)(unsigned)(size_t)l;
  __builtin_amdgcn_global_load_async_to_lds_b128(gp, lp, 0, 0);
  __builtin_amdgcn_global_load_async_to_lds_b128(gp, lp, 16, 0);
  __builtin_amdgcn_global_load_async_to_lds_b128(gp, lp, 32, 0);
  __builtin_amdgcn_global_load_async_to_lds_b128(gp, lp, 48, 0);
#else
  const float4* g4 = (const float4*)g;
  float4* l4 = (float4*)l;
#pragma unroll
  for (int k = 0; k < 4; ++k) l4[k] = g4[k];
#endif
}

// ---------------- Kernel 1: srcsum[b][d] = sum_s w_source[b][s][d] ----------------
__global__ __launch_bounds__(128) void ma_srcsum_kernel(const float* __restrict__ src,
                                                        float* __restrict__ srcsum) {
  const int b = blockIdx.x;
  const int d = threadIdx.x;  // 0..127 == D
  const float* p = src + (size_t)b * S * D + d;
  float acc = 0.f;
#pragma unroll 8
  for (int s = 0; s < S; ++s) acc += p[(size_t)s * D];
  srcsum[b * D + d] = acc;
}

// ------------- Kernel 2: colsum[b][t] = <srcsum[b], w_target[b][t]> ---------------
__global__ __launch_bounds__(256) void ma_colsum_kernel(const float* __restrict__ tgt,
                                                        const float* __restrict__ srcsum,
                                                        float* __restrict__ colsum) {
  __shared__ float ssum[D];
  const int b = blockIdx.y;
  const int t = blockIdx.x * 256 + threadIdx.x;
  if (threadIdx.x < D) ssum[threadIdx.x] = srcsum[b * D + threadIdx.x];
  __syncthreads();
  const float4* row = (const float4*)(tgt + ((size_t)b * T + t) * D);
  float acc = 0.f;
#pragma unroll
  for (int k = 0; k < D / 4; ++k) {
    float4 v = row[k];
    float4 s = ((const float4*)ssum)[k];
    acc += v.x * s.x + v.y * s.y + v.z * s.z + v.w * s.w;
  }
  colsum[b * T + t] = acc;
}

// ------- Kernel 3: fused WMMA GEMM + (colsum - score) argmax + row gather --------
__global__ __launch_bounds__(128) void ma_argmax_gather_kernel(const float* __restrict__ src,
                                                               const float* __restrict__ tgt,
                                                               const float* __restrict__ colsum,
                                                               float* __restrict__ out) {
  __shared__ float tile[2][16 * LDSW];  // double-buffered 16x128 w_target tiles
  __shared__ int widx[4 * 16];          // winning t per row, per wave

  const int b = blockIdx.y;
  const int tid = threadIdx.x;
  const int wave = tid >> 5;
  const int lane = tid & 31;
  const int l16 = lane & 15;
  const int hi = lane >> 4;  // 0 or 1 (lane half)
  const int s0 = blockIdx.x * 64 + wave * 16;

  const float* srcB = src + (size_t)b * S * D;
  const float* tgtB = tgt + (size_t)b * T * D;
  const float* csB = colsum + (size_t)b * T;

  // A-matrix (16x4 fp32 per WMMA): lane holds (m = l16, k = 4*kb + 2*hi + {0,1}).
  // Keep all K=128 in registers: 32 x v2f, each a contiguous 8B global load.
  v2f areg[32];
  const float* arow = srcB + (size_t)(s0 + l16) * D + 2 * hi;
#pragma unroll
  for (int kb = 0; kb < 32; ++kb) areg[kb] = *(const v2f*)(arow + kb * 4);

  const float NEG_INF = -__builtin_inff();
  float bestv[8];
  int besti[8];
#pragma unroll
  for (int r = 0; r < 8; ++r) {
    bestv[r] = NEG_INF;
    besti[r] = 0x7fffffff;
  }

  // Prologue: stage tile 0, then steady state: issue async for tile n+1,
  // compute WMMAs on tile n, wait + one barrier per iteration.
  ma_stage_tile(tile[0], tgtB, tid);
  ma_wait_async();
  __syncthreads();

  int cur = 0;
  for (int t0 = 0; t0 < T; t0 += 16) {
    if (t0 + 16 < T) ma_stage_tile(tile[cur ^ 1], tgtB + (size_t)(t0 + 16) * D, tid);

    // B-matrix (4x16 fp32): lane holds (n = l16, k = 4*kb + 2*hi + {0,1}),
    // i.e. tgt row (t0 + l16). Conflict-free ds_load_b64 from padded LDS.
    // Two independent accumulator chains (even/odd kb) for matrix-pipe ILP.
    v8f acc0 = {0.f, 0.f, 0.f, 0.f, 0.f, 0.f, 0.f, 0.f};
    v8f acc1 = {0.f, 0.f, 0.f, 0.f, 0.f, 0.f, 0.f, 0.f};
    const float* brow = &tile[cur][l16 * LDSW + 2 * hi];
#pragma unroll
    for (int kb = 0; kb < 32; kb += 2) {
      v2f b0 = *(const v2f*)(brow + kb * 4);
      v2f b1 = *(const v2f*)(brow + kb * 4 + 4);
      acc0 = __builtin_amdgcn_wmma_f32_16x16x4_f32(false, areg[kb], false, b0,
                                                   (short)0, acc0, false, false);
      acc1 = __builtin_amdgcn_wmma_f32_16x16x4_f32(false, areg[kb + 1], false, b1,
                                                   (short)0, acc1, false, false);
    }

    // D layout: VGPR r, lane -> (row = r + 8*hi, col t = t0 + l16).
    const int t = t0 + l16;
    const float ct = csB[t];
#pragma unroll
    for (int r = 0; r < 8; ++r) {
      float left = ct - (acc0[r] + acc1[r]);
      if (left > bestv[r]) {  // per-lane t is increasing: '>' keeps first occurrence
        bestv[r] = left;
        besti[r] = t;
      }
    }

    ma_wait_async();
    __syncthreads();
    cur ^= 1;
  }

  // Reduce argmax across the 16 column-lanes of each half-wave (first-index ties).
#pragma unroll
  for (int r = 0; r < 8; ++r) {
    float v = bestv[r];
    int i = besti[r];
#pragma unroll
    for (int m = 8; m >= 1; m >>= 1) {
      float ov = __shfl_xor(v, m, 32);
      int oi = __shfl_xor(i, m, 32);
      if (ov > v || (ov == v && oi < i)) {
        v = ov;
        i = oi;
      }
    }
    bestv[r] = v;
    besti[r] = i;
  }
  if (l16 == 0) {
#pragma unroll
    for (int r = 0; r < 8; ++r) widx[wave * 16 + hi * 8 + r] = besti[r];
  }
  __syncthreads();

  // Gather: out[b][s0+row][:] = tgt[b][widx[row]][:]; 32 lanes x float4 = 128 floats.
#pragma unroll 1
  for (int row = 0; row < 16; ++row) {
    const int idx = widx[wave * 16 + row];
    const float4* g = (const float4*)(tgtB + (size_t)idx * D);
    float4* o = (float4*)(out + ((size_t)b * S + s0 + row) * D);
    o[lane] = g[lane];
  }
}

extern "C" void kernel_launch(void* const* d_in, const int* in_sizes, int n_in,
                              void* d_out, int out_size, void* d_ws, size_t ws_size,
                              hipStream_t stream) {
  (void)in_sizes;
  (void)n_in;
  (void)out_size;
  (void)ws_size;
  const float* src = (const float*)d_in[0];  // [B,S,D]
  const float* tgt = (const float*)d_in[1];  // [B,T,D]
  float* out = (float*)d_out;                // [B,S,D]

  float* ws = (float*)d_ws;
  float* srcsum = ws;          // B*D floats
  float* colsum = ws + B * D;  // B*T floats

  ma_srcsum_kernel<<<dim3(B), 128, 0, stream>>>(src, srcsum);
  ma_colsum_kernel<<<dim3(T / 256, B), 256, 0, stream>>>(tgt, srcsum, colsum);
  ma_argmax_gather_kernel<<<dim3(S / 64, B), 128, 0, stream>>>(src, tgt, colsum, out);
}